// HJRL_67619965108617
// MI455X (gfx1250) — compile-verified
//
#include <hip/hip_runtime.h>

#define NN 50000
#define MM 20000
#define FIN 256
#define HID 256
#define OUTD 64
#define NEG_SLOPE 0.2f

typedef __attribute__((ext_vector_type(2))) float v2f;
typedef __attribute__((ext_vector_type(8))) float v8f;

__device__ __forceinline__ float leaky_f(float v) {
    return v >= 0.0f ? v : NEG_SLOPE * v;
}

// ---------------------------------------------------------------------------
// Zero fill, float4-vectorized (n must be a multiple of 4 — true for all uses)
// ---------------------------------------------------------------------------
__global__ void fill_zero_kernel(float4* __restrict__ p, size_t n4) {
    size_t i = (size_t)blockIdx.x * blockDim.x + threadIdx.x;
    size_t stride = (size_t)gridDim.x * blockDim.x;
    const float4 z = make_float4(0.f, 0.f, 0.f, 0.f);
    for (; i < n4; i += stride) p[i] = z;
}

// ---------------------------------------------------------------------------
// dst[i] = leaky(a[i]) + leaky(b[i])   (dst may alias a), float4-vectorized
// ---------------------------------------------------------------------------
__global__ void combine_leaky_kernel(float4* __restrict__ dst,
                                     const float4* __restrict__ a,
                                     const float4* __restrict__ b, size_t n4) {
    size_t i = (size_t)blockIdx.x * blockDim.x + threadIdx.x;
    size_t stride = (size_t)gridDim.x * blockDim.x;
    for (; i < n4; i += stride) {
        float4 av = a[i], bv = b[i], r;
        r.x = leaky_f(av.x) + leaky_f(bv.x);
        r.y = leaky_f(av.y) + leaky_f(bv.y);
        r.z = leaky_f(av.z) + leaky_f(bv.z);
        r.w = leaky_f(av.w) + leaky_f(bv.w);
        dst[i] = r;
    }
}

// ---------------------------------------------------------------------------
// C[rows x out_cols] = A[rows x 256] @ W[256 x out_cols]
// One wave32 computes a 32x64 output tile: 8 accumulators, 8 WMMAs per
// k-step of 4. The 4 B-fragments are shared by both 16-row groups
// (B traffic /2 vs 16x64), and each A-fragment feeds 4 column tiles.
// Fragment layouts per CDNA5 ISA (V_WMMA_F32_16X16X4_F32):
//   A (16x4): lanes 0-15 -> M=lane, {K0,K1}; lanes 16-31 -> {K2,K3}
//   B (4x16): VGPR0 lanes0-15 = row K0, lanes16-31 = row K2; VGPR1: K1/K3
//   C/D: lane<16 -> M=r, N=lane; lane>=16 -> M=r+8, N=lane-16
// rows % 16 == 0 (each 16-row group is fully valid or fully absent);
// out_cols % 64 == 0.
// ---------------------------------------------------------------------------
__global__ void gemm_wmma_f32_kernel(const float* __restrict__ A,
                                     const float* __restrict__ W,
                                     float* __restrict__ C,
                                     int rows, int out_cols) {
    const int lane = (int)(threadIdx.x & 31);
    const int half = lane >> 4;      // 0: K pair {0,1}, 1: K pair {2,3}
    const int l    = lane & 15;
    const int col0 = blockIdx.x * 64;
    const int row0 = blockIdx.y * 32;

    const bool grp1 = (row0 + 16) < rows;          // second 16-row group present?
    const int r0 = row0 + l;                       // group-0 row for this lane
    const int r1 = (grp1 ? row0 + 16 : row0) + l;  // clamp loads when absent

    v8f c0 = {}, c1 = {}, c2 = {}, c3 = {};        // rows [row0, row0+16)
    v8f d0 = {}, d1 = {}, d2 = {}, d3 = {};        // rows [row0+16, row0+32)

    const float* ap0 = A + (size_t)r0 * FIN + half * 2;
    const float* ap1 = A + (size_t)r1 * FIN + half * 2;
    const float* wp  = W + (size_t)(half * 2) * out_cols + col0 + l;

    for (int k = 0; k < FIN; k += 4) {
        v2f a0, a1;
        a0.x = ap0[0];
        a0.y = ap0[1];
        a1.x = ap1[0];
        a1.y = ap1[1];
        ap0 += 4;
        ap1 += 4;
        v2f b0, b1, b2, b3;
        b0.x = wp[0];
        b0.y = wp[out_cols];
        b1.x = wp[16];
        b1.y = wp[out_cols + 16];
        b2.x = wp[32];
        b2.y = wp[out_cols + 32];
        b3.x = wp[48];
        b3.y = wp[out_cols + 48];
        wp += (size_t)4 * out_cols;
        // (neg_a, A, neg_b, B, c_mod, C, reuse_a, reuse_b)
        c0 = __builtin_amdgcn_wmma_f32_16x16x4_f32(false, a0, false, b0, (short)0, c0, false, false);
        c1 = __builtin_amdgcn_wmma_f32_16x16x4_f32(false, a0, false, b1, (short)0, c1, false, false);
        c2 = __builtin_amdgcn_wmma_f32_16x16x4_f32(false, a0, false, b2, (short)0, c2, false, false);
        c3 = __builtin_amdgcn_wmma_f32_16x16x4_f32(false, a0, false, b3, (short)0, c3, false, false);
        d0 = __builtin_amdgcn_wmma_f32_16x16x4_f32(false, a1, false, b0, (short)0, d0, false, false);
        d1 = __builtin_amdgcn_wmma_f32_16x16x4_f32(false, a1, false, b1, (short)0, d1, false, false);
        d2 = __builtin_amdgcn_wmma_f32_16x16x4_f32(false, a1, false, b2, (short)0, d2, false, false);
        d3 = __builtin_amdgcn_wmma_f32_16x16x4_f32(false, a1, false, b3, (short)0, d3, false, false);
    }

    float* cp = C + (size_t)(row0 + half * 8) * out_cols + col0 + l;
#pragma unroll
    for (int r = 0; r < 8; ++r) {
        const size_t o = (size_t)r * out_cols;
        cp[o]      = c0[r];
        cp[o + 16] = c1[r];
        cp[o + 32] = c2[r];
        cp[o + 48] = c3[r];
    }
    if (grp1) {
        float* dp = C + (size_t)(row0 + 16 + half * 8) * out_cols + col0 + l;
#pragma unroll
        for (int r = 0; r < 8; ++r) {
            const size_t o = (size_t)r * out_cols;
            dp[o]      = d0[r];
            dp[o + 16] = d1[r];
            dp[o + 32] = d2[r];
            dp[o + 48] = d3[r];
        }
    }
}

// ---------------------------------------------------------------------------
// COO SpMM with atomic scatter: out[rows[e], :] += vals[e] * dense[cols[e], :]
// D/4 threads cooperate per nonzero, each handling a float4 chunk.
// ---------------------------------------------------------------------------
template <int D>
__global__ void spmm_atomic_kernel(const int* __restrict__ rows,
                                   const int* __restrict__ cols,
                                   const float* __restrict__ vals,
                                   const float* __restrict__ dense,
                                   float* __restrict__ out, int nnz) {
    constexpr int TPE = D / 4;        // threads per edge
    constexpr int EPB = 256 / TPE;    // edges per 256-thread block
    const int e = blockIdx.x * EPB + (int)(threadIdx.x / TPE);
    if (e >= nnz) return;
    const int t = (int)(threadIdx.x % TPE);
    const int r = rows[e];
    const int ci = cols[e];
    const float v = vals[e];
    const float4 d4 = ((const float4*)(dense + (size_t)ci * D))[t];
    float* op = out + (size_t)r * D + t * 4;
    atomicAdd(op + 0, v * d4.x);
    atomicAdd(op + 1, v * d4.y);
    atomicAdd(op + 2, v * d4.z);
    atomicAdd(op + 3, v * d4.w);
}

// ---------------------------------------------------------------------------
extern "C" void kernel_launch(void* const* d_in, const int* in_sizes, int n_in,
                              void* d_out, int out_size, void* d_ws, size_t ws_size,
                              hipStream_t stream) {
    (void)n_in; (void)out_size; (void)ws_size;

    const float* x  = (const float*)d_in[0];
    const float* y  = (const float*)d_in[1];
    const float* W0 = (const float*)d_in[2];
    const float* W1 = (const float*)d_in[3];
    const int*   hht_r = (const int*)d_in[4];
    const int*   hht_c = (const int*)d_in[5];
    const float* hht_v = (const float*)d_in[6];
    const int*   h_r   = (const int*)d_in[7];
    const int*   h_c   = (const int*)d_in[8];
    const float* h_v   = (const float*)d_in[9];
    const int*   ht_r  = (const int*)d_in[10];
    const int*   ht_c  = (const int*)d_in[11];
    const float* ht_v  = (const float*)d_in[12];
    const int*   hth_r = (const int*)d_in[13];
    const int*   hth_c = (const int*)d_in[14];
    const float* hth_v = (const float*)d_in[15];

    const int nnz_hht = in_sizes[4];
    const int nnz_h   = in_sizes[7];
    const int nnz_ht  = in_sizes[10];
    const int nnz_hth = in_sizes[13];

    // Workspace layout (floats)
    float* ws    = (float*)d_ws;
    float* bufXW = ws;                            // NN*HID  (x@W0; later x1@W1 NN*64)
    float* bufYW = bufXW + (size_t)NN * HID;      // MM*HID  (y@W0; later y1@W1 MM*64)
    float* accA  = bufYW + (size_t)MM * HID;      // NN*HID  (hht partial -> x1)
    float* accB  = accA  + (size_t)NN * HID;      // NN*HID  (h partial; later layer1 N-accs)
    float* accC  = accB  + (size_t)NN * HID;      // MM*HID  (ht partial -> y1)
    float* accD  = accC  + (size_t)MM * HID;      // MM*HID  (hth partial; later layer1 M-accs)

    float* out_x = (float*)d_out;                 // NN*OUTD
    float* out_y = out_x + (size_t)NN * OUTD;     // MM*OUTD

    const int gyN = (NN + 31) / 32;               // 1563 (last block: 16-row group only)
    const int gyM = (MM + 31) / 32;               // 625

    // ---- zero the layer-0 accumulators (accA..accD are contiguous) ----
    const size_t zn0 = 2 * ((size_t)NN + (size_t)MM) * HID;
    fill_zero_kernel<<<2048, 256, 0, stream>>>((float4*)accA, zn0 / 4);

    // ---- layer 0 dense GEMMs: xW0, yW0 (fp32 WMMA, 32x64 tiles) ----
    gemm_wmma_f32_kernel<<<dim3(HID / 64, gyN), 32, 0, stream>>>(x, W0, bufXW, NN, HID);
    gemm_wmma_f32_kernel<<<dim3(HID / 64, gyM), 32, 0, stream>>>(y, W0, bufYW, MM, HID);

    // ---- layer 0 SpMMs (D = 256) ----
    {
        const int epb = 4; // 256 threads / (256/4)
        spmm_atomic_kernel<HID><<<(nnz_hht + epb - 1) / epb, 256, 0, stream>>>(
            hht_r, hht_c, hht_v, bufXW, accA, nnz_hht);
        spmm_atomic_kernel<HID><<<(nnz_h + epb - 1) / epb, 256, 0, stream>>>(
            h_r, h_c, h_v, bufYW, accB, nnz_h);
        spmm_atomic_kernel<HID><<<(nnz_ht + epb - 1) / epb, 256, 0, stream>>>(
            ht_r, ht_c, ht_v, bufXW, accC, nnz_ht);
        spmm_atomic_kernel<HID><<<(nnz_hth + epb - 1) / epb, 256, 0, stream>>>(
            hth_r, hth_c, hth_v, bufYW, accD, nnz_hth);
    }

    // ---- x1 = leaky(accA)+leaky(accB) (in-place accA); y1 likewise in accC ----
    combine_leaky_kernel<<<4096, 256, 0, stream>>>(
        (float4*)accA, (const float4*)accA, (const float4*)accB, (size_t)NN * HID / 4);
    combine_leaky_kernel<<<4096, 256, 0, stream>>>(
        (float4*)accC, (const float4*)accC, (const float4*)accD, (size_t)MM * HID / 4);

    // ---- layer 1 dense GEMMs: x1@W1 -> bufXW (NN*64), y1@W1 -> bufYW (MM*64) ----
    gemm_wmma_f32_kernel<<<dim3(OUTD / 64, gyN), 32, 0, stream>>>(accA, W1, bufXW, NN, OUTD);
    gemm_wmma_f32_kernel<<<dim3(OUTD / 64, gyM), 32, 0, stream>>>(accC, W1, bufYW, MM, OUTD);

    // ---- layer 1 accumulators (reuse accB / accD space) ----
    float* p_hht = accB;                          // NN*OUTD
    float* p_h   = accB + (size_t)NN * OUTD;      // NN*OUTD
    float* p_ht  = accD;                          // MM*OUTD
    float* p_hth = accD + (size_t)MM * OUTD;      // MM*OUTD
    fill_zero_kernel<<<1024, 256, 0, stream>>>((float4*)accB, 2 * (size_t)NN * OUTD / 4);
    fill_zero_kernel<<<1024, 256, 0, stream>>>((float4*)accD, 2 * (size_t)MM * OUTD / 4);

    // ---- layer 1 SpMMs (D = 64) ----
    {
        const int epb = 16; // 256 threads / (64/4)
        spmm_atomic_kernel<OUTD><<<(nnz_hht + epb - 1) / epb, 256, 0, stream>>>(
            hht_r, hht_c, hht_v, bufXW, p_hht, nnz_hht);
        spmm_atomic_kernel<OUTD><<<(nnz_h + epb - 1) / epb, 256, 0, stream>>>(
            h_r, h_c, h_v, bufYW, p_h, nnz_h);
        spmm_atomic_kernel<OUTD><<<(nnz_ht + epb - 1) / epb, 256, 0, stream>>>(
            ht_r, ht_c, ht_v, bufXW, p_ht, nnz_ht);
        spmm_atomic_kernel<OUTD><<<(nnz_hth + epb - 1) / epb, 256, 0, stream>>>(
            hth_r, hth_c, hth_v, bufYW, p_hth, nnz_hth);
    }

    // ---- final combines into d_out ----
    combine_leaky_kernel<<<2048, 256, 0, stream>>>(
        (float4*)out_x, (const float4*)p_hht, (const float4*)p_h, (size_t)NN * OUTD / 4);
    combine_leaky_kernel<<<2048, 256, 0, stream>>>(
        (float4*)out_y, (const float4*)p_ht, (const float4*)p_hth, (size_t)MM * OUTD / 4);
}